// GCN_1597727834289
// MI455X (gfx1250) — compile-verified
//
#include <hip/hip_runtime.h>
#include <hip/hip_bf16.h>

// ---------------------------------------------------------------------------
// GCN on MI455X (gfx1250):
//   - fp32 WMMA (V_WMMA_F32_16X16X4_F32) for the per-layer GEMM
//   - CSR-by-target gather aggregation (no float atomics in the hot loop);
//     norm + self-loop + bias + tanh fused into one pass
// ---------------------------------------------------------------------------

typedef float v2f __attribute__((ext_vector_type(2)));
typedef float v8f __attribute__((ext_vector_type(8)));

#define NUM_GRAPHS 512
#define D 64

// ---------------------------------------------------------------------------
// GEMM: out[N x 64] = A[N x fin] @ W[fin x 64], fp32 WMMA 16x16x4.
// One wave per 16x16 output tile. nrow % 16 == 0, fin % 4 == 0.
// A-layout (ISA 7.12.2, 32-bit A 16x4): lane l holds A[l%16, k0 + 2*(l/16) + {0,1}]
// B-layout (row-striped):               lane l holds B[k0 + 2*(l/16) + {0,1}, l%16]
// C-layout: vgpr r of lane l = C[r + 8*(l/16), l%16]
// ---------------------------------------------------------------------------
__global__ void gemm_wmma_f32(const float* __restrict__ A,
                              const float* __restrict__ W,
                              float* __restrict__ out,
                              int nrow, int fin) {
    const int wavesPerBlock = blockDim.x >> 5;
    const int wave = blockIdx.x * wavesPerBlock + (threadIdx.x >> 5);
    const int lane = threadIdx.x & 31;
    const int tilesM = nrow >> 4;
    const int totalTiles = tilesM * (D / 16);
    if (wave >= totalTiles) return;             // wave-uniform: EXEC stays all-1s

    const int tm = wave >> 2;                   // D/16 == 4 column tiles
    const int tn = wave & 3;
    const int half = lane >> 4;                 // 0: K pair {0,1}; 1: K pair {2,3}
    const int m = (tm << 4) + (lane & 15);
    const int n = (tn << 4) + (lane & 15);

    const float* arow = A + (long long)m * fin;
    __builtin_prefetch(arow, 0, 3);             // global_prefetch_b8

    v8f c = {};
    for (int k0 = 0; k0 < fin; k0 += 4) {
        const int ka = k0 + (half << 1);
        v2f a = *(const v2f*)(arow + ka);       // A[m, ka], A[m, ka+1]
        v2f b;
        b.x = W[(long long)ka * D + n];         // B[ka,   n]
        b.y = W[(long long)(ka + 1) * D + n];   // B[ka+1, n]
        c = __builtin_amdgcn_wmma_f32_16x16x4_f32(
                /*neg_a=*/false, a, /*neg_b=*/false, b,
                /*c_mod=*/(short)0, c, /*reuse_a=*/false, /*reuse_b=*/false);
    }

    float* orow = out + (long long)((tm << 4) + (half << 3)) * D + n;
#pragma unroll
    for (int r = 0; r < 8; ++r)
        orow[(long long)r * D] = c[r];
}

// ---------------------------------------------------------------------------
// CSR construction (by target node) + normalization
// ---------------------------------------------------------------------------
__global__ void init_counts(int* __restrict__ counts, int N) {
    int i = blockIdx.x * blockDim.x + threadIdx.x;
    if (i < N) counts[i] = 0;
}

__global__ void count_in(const int* __restrict__ col, int* __restrict__ counts, int E) {
    int e = blockIdx.x * blockDim.x + threadIdx.x;
    if (e < E) atomicAdd(&counts[col[e]], 1);   // native int atomic
}

__global__ void compute_dinv(const int* __restrict__ counts, float* __restrict__ dinv, int N) {
    int i = blockIdx.x * blockDim.x + threadIdx.x;
    if (i < N) dinv[i] = rsqrtf((float)(counts[i] + 1));  // +1 self-loop
}

// Single-block chunked exclusive scan over counts -> offsets[0..N]
__global__ void scan_counts(const int* __restrict__ counts, int* __restrict__ offsets, int N) {
    __shared__ int sh[1024];
    __shared__ int carry;
    if (threadIdx.x == 0) carry = 0;
    __syncthreads();
    for (int base = 0; base < N; base += 1024) {
        int i = base + (int)threadIdx.x;
        int v = (i < N) ? counts[i] : 0;
        sh[threadIdx.x] = v;
        __syncthreads();
        // Hillis-Steele inclusive scan in LDS
        for (int off = 1; off < 1024; off <<= 1) {
            int t = (threadIdx.x >= (unsigned)off) ? sh[threadIdx.x - off] : 0;
            __syncthreads();
            sh[threadIdx.x] += t;
            __syncthreads();
        }
        int incl = sh[threadIdx.x];
        int c = carry;                          // everyone reads current carry
        __syncthreads();                        // before thread 0 bumps it
        if (threadIdx.x == 0) carry = c + sh[1023];
        if (i < N) offsets[i] = c + incl - v;   // exclusive
        __syncthreads();
    }
    if (threadIdx.x == 0) offsets[N] = carry;   // == E
}

__global__ void copy_cursor(const int* __restrict__ offsets, int* __restrict__ cursor, int N) {
    int i = blockIdx.x * blockDim.x + threadIdx.x;
    if (i < N) cursor[i] = offsets[i];
}

__global__ void fill_csr(const int* __restrict__ row, const int* __restrict__ col,
                         int* __restrict__ cursor, int* __restrict__ srcs, int E) {
    int e = blockIdx.x * blockDim.x + threadIdx.x;
    if (e >= E) return;
    int c = col[e];
    int p = atomicAdd(&cursor[c], 1);           // native int atomic
    srcs[p] = row[e];
}

// ---------------------------------------------------------------------------
// Fused aggregation: one wave per node, 32 lanes x float2 = 64 features.
// out[i] = tanh( dinv[i] * ( tmp[i]*dinv[i] + sum_{r in in(i)} tmp[r]*dinv[r] ) + b )
// Pure gathers from L2-resident tmp; no atomics.
// ---------------------------------------------------------------------------
__global__ void aggregate(const float* __restrict__ tmp, const float* __restrict__ dinv,
                          const int* __restrict__ offsets, const int* __restrict__ srcs,
                          const float* __restrict__ bias, float* __restrict__ hout, int N) {
    const int wavesPerBlock = blockDim.x >> 5;
    const int node = blockIdx.x * wavesPerBlock + (threadIdx.x >> 5);
    if (node >= N) return;
    const int lane = threadIdx.x & 31;
    const int f = lane << 1;

    const int s = offsets[node];
    const int e = offsets[node + 1];
    const float di = dinv[node];

    // self-loop term (pre final *di)
    v2f acc = *(const v2f*)(tmp + (long long)node * D + f);
    acc.x *= di; acc.y *= di;

    for (int p = s; p < e; ++p) {
        int r = srcs[p];                        // uniform across wave -> 1 L2 req
        float dr = dinv[r];
        v2f v = *(const v2f*)(tmp + (long long)r * D + f);
        acc.x += v.x * dr;
        acc.y += v.y * dr;
    }

    v2f o;
    o.x = tanhf(acc.x * di + bias[f]);
    o.y = tanhf(acc.y * di + bias[f + 1]);
    *(v2f*)(hout + (long long)node * D + f) = o;
}

// ---------------------------------------------------------------------------
// Per-graph segment bounds (batch_index is sorted -> contiguous segments)
// ---------------------------------------------------------------------------
__global__ void init_bounds(int* __restrict__ gstart, int* __restrict__ gend, int N) {
    int g = blockIdx.x * blockDim.x + threadIdx.x;
    if (g < NUM_GRAPHS) { gstart[g] = N; gend[g] = 0; }
}

__global__ void compute_bounds(const int* __restrict__ batch,
                               int* __restrict__ gstart, int* __restrict__ gend, int N) {
    int i = blockIdx.x * blockDim.x + threadIdx.x;
    if (i >= N) return;
    int g = batch[i];
    atomicMin(&gstart[g], i);
    atomicMax(&gend[g], i + 1);
}

// ---------------------------------------------------------------------------
// Pooling + output head. One 64-thread block per graph.
// out layout: [0,512) = out, [512, 512+512*128) = hidden (max||mean).
// ---------------------------------------------------------------------------
__global__ void pool_out(const float* __restrict__ h,
                         const int* __restrict__ gstart, const int* __restrict__ gend,
                         const float* __restrict__ Wout, const float* __restrict__ bout,
                         float* __restrict__ out) {
    int g = blockIdx.x;
    int f = threadIdx.x;                        // 0..63
    int s = gstart[g], e = gend[g];

    float m = -INFINITY, sum = 0.0f;
    for (int i = s; i < e; ++i) {
        float v = h[(long long)i * D + f];
        m = fmaxf(m, v);
        sum += v;
    }
    float mean;
    if (e <= s) { m = 0.0f; mean = 0.0f; }      // empty graph -> 0 (matches reference)
    else        { mean = sum / (float)(e - s); }

    out[NUM_GRAPHS + (long long)g * (2 * D) + f]     = m;
    out[NUM_GRAPHS + (long long)g * (2 * D) + D + f] = mean;

    __shared__ float red[D];
    red[f] = m * Wout[f] + mean * Wout[D + f];
    __syncthreads();
    if (f == 0) {
        float acc = bout[0];
#pragma unroll
        for (int j = 0; j < D; ++j) acc += red[j];
        out[g] = acc;
    }
}

// ---------------------------------------------------------------------------
// Launcher
// ---------------------------------------------------------------------------
static inline int cdiv(long long a, long long b) { return (int)((a + b - 1) / b); }

extern "C" void kernel_launch(void* const* d_in, const int* in_sizes, int n_in,
                              void* d_out, int out_size, void* d_ws, size_t ws_size,
                              hipStream_t stream) {
    const float* x     = (const float*)d_in[0];
    const int*   eidx  = (const int*)  d_in[1];
    const int*   batch = (const int*)  d_in[2];
    const float* W0    = (const float*)d_in[3];
    const float* b0    = (const float*)d_in[4];
    const float* W1    = (const float*)d_in[5];
    const float* b1    = (const float*)d_in[6];
    const float* W2    = (const float*)d_in[7];
    const float* b2    = (const float*)d_in[8];
    const float* W3    = (const float*)d_in[9];
    const float* b3    = (const float*)d_in[10];
    const float* Wout  = (const float*)d_in[11];
    const float* bout  = (const float*)d_in[12];

    const int N = in_sizes[0] / 128;            // 50000
    const int E = in_sizes[1] / 2;              // 800000
    const int* row = eidx;                      // edge_index[0] = source
    const int* col = eidx + E;                  // edge_index[1] = target

    // Workspace carve-up (~42 MB)
    float* tmp    = (float*)d_ws;               // [N*64] post-GEMM
    float* hA     = tmp + (size_t)N * D;        // [N*64]
    float* hB     = hA  + (size_t)N * D;        // [N*64]
    float* dinv   = hB  + (size_t)N * D;        // [N]
    int* counts   = (int*)(dinv + N);           // [N]
    int* offsets  = counts + N;                 // [N+1]
    int* cursor   = offsets + N + 1;            // [N]
    int* srcs     = cursor + N;                 // [E]
    int* gstart   = srcs + E;                   // [512]
    int* gend     = gstart + NUM_GRAPHS;        // [512]

    const int T = 256;

    // --- CSR by target + normalization (rebuilt every call; ws is not persistent)
    init_counts <<<cdiv(N, T), T, 0, stream>>>(counts, N);
    count_in    <<<cdiv(E, T), T, 0, stream>>>(col, counts, E);
    compute_dinv<<<cdiv(N, T), T, 0, stream>>>(counts, dinv, N);
    scan_counts <<<1, 1024, 0, stream>>>(counts, offsets, N);
    copy_cursor <<<cdiv(N, T), T, 0, stream>>>(offsets, cursor, N);
    fill_csr    <<<cdiv(E, T), T, 0, stream>>>(row, col, cursor, srcs, E);

    // --- 4 GCN layers
    const float* Ws[4] = {W0, W1, W2, W3};
    const float* bs[4] = {b0, b1, b2, b3};
    const float* hin = x;
    int fin = 128;
    float* cur = hA;
    float* nxt = hB;

    const int tiles = (N / 16) * (D / 16);      // one wave per 16x16 tile
    const int gemmBlocks = cdiv(tiles, T / 32);
    const int aggBlocks  = cdiv(N, T / 32);     // one wave per node

    for (int l = 0; l < 4; ++l) {
        gemm_wmma_f32<<<gemmBlocks, T, 0, stream>>>(hin, Ws[l], tmp, N, fin);
        aggregate    <<<aggBlocks,  T, 0, stream>>>(tmp, dinv, offsets, srcs, bs[l], cur, N);
        hin = cur; fin = D;
        float* t = cur; cur = nxt; nxt = t;
    }

    // --- Pooling + head
    init_bounds   <<<cdiv(NUM_GRAPHS, T), T, 0, stream>>>(gstart, gend, N);
    compute_bounds<<<cdiv(N, T), T, 0, stream>>>(batch, gstart, gend, N);
    pool_out      <<<NUM_GRAPHS, D, 0, stream>>>(hin, gstart, gend, Wout, bout, (float*)d_out);
}